// ComputeNodeAreaFromPinMap_12816182411854
// MI455X (gfx1250) — compile-verified
//
#include <hip/hip_runtime.h>

// ComputeNodeAreaFromPinMap for MI455X (gfx1250, wave32).
//
// Memory-bound elementwise + gather kernel:
//   per node: 24 B streamed (NT hints) + 4 gathers into a 1 MB L2-resident map.
// Roofline floor ~2 us at 23.3 TB/s. No matmul structure -> WMMA not applicable.
//
// Numerics policy:
//  - floor(x/BS) uses true correctly-rounded fp32 division (bit-exact bin
//    indices vs the JAX reference; an approximate reciprocal could flip
//    borderline bins).
//  - final scale 1/(w*h*0.1) uses v_rcp_f32 + mul (~1 ulp): pure scaling,
//    no integer decision, ~2e-7 relative error; removes 4 precise-division
//    sequences (~36 VALU ops + serial chains) per thread.

// clang native vectors (HIP's float4/int4 are structs and are rejected by
// __builtin_nontemporal_load/store).
typedef float v4f __attribute__((ext_vector_type(4)));
typedef int   v4i __attribute__((ext_vector_type(4)));

namespace {
constexpr int   kN          = 2000000;          // nodes
constexpr int   kNB         = 512;              // NBX == NBY
constexpr float kBS         = 1000.0f / 512.0f; // BSX == BSY (exact in fp32)
constexpr float kUnitPinCap = 0.1f;
constexpr int   kVec        = 4;                // nodes per thread (b128 loads)
constexpr int   kThreads    = 256;              // 8 wave32s per block
}

// Overlap of [lo, lo+len) with bins b0 and b0+1 (b0 = floor(lo/BS)),
// exactly mirroring the reference: bin_lo = b*BS (unclipped), overlap
// clamped below at 0, index clamped to [0, 511].
__device__ __forceinline__ void axis_overlap(float lo, float len,
                                             float& ov0, float& ov1,
                                             int& i0, int& i1) {
  const float hi  = lo + len;
  const float b0f = floorf(lo / kBS);      // true fp32 divide (matches ref)
  const int   b0  = (int)b0f;
  const float bl0 = b0f * kBS;             // exact: b*1000/512, ulp-free
  const float bl1 = bl0 + kBS;
  ov0 = fmaxf(fminf(hi, bl0 + kBS) - fmaxf(lo, bl0), 0.0f);
  ov1 = fmaxf(fminf(hi, bl1 + kBS) - fmaxf(lo, bl1), 0.0f);
  i0 = min(max(b0,     0), kNB - 1);
  i1 = min(max(b0 + 1, 0), kNB - 1);
}

__global__ __launch_bounds__(kThreads)
void node_area_from_pinmap_kernel(const float* __restrict__ pos,        // [2N] x then y
                                  const float* __restrict__ node_size_x,// [N]
                                  const float* __restrict__ node_size_y,// [N]
                                  const float* __restrict__ util,       // [512*512], C-order
                                  const int*  __restrict__ pin_w,       // [N]
                                  float*      __restrict__ out)         // [N]
{
  const int vid  = blockIdx.x * kThreads + threadIdx.x;
  const int base = vid * kVec;
  if (base >= kN) return;   // kN % 4 == 0, so base+3 < kN whenever base < kN

  // Streaming inputs: touched once -> non-temporal (th:NT) so the hot 1 MB
  // utilization map keeps L2 priority. All b128, fully coalesced.
  const v4f x4 = __builtin_nontemporal_load(reinterpret_cast<const v4f*>(pos + base));
  const v4f y4 = __builtin_nontemporal_load(reinterpret_cast<const v4f*>(pos + kN + base));
  const v4f w4 = __builtin_nontemporal_load(reinterpret_cast<const v4f*>(node_size_x + base));
  const v4f h4 = __builtin_nontemporal_load(reinterpret_cast<const v4f*>(node_size_y + base));
  const v4i p4 = __builtin_nontemporal_load(reinterpret_cast<const v4i*>(pin_w + base));

  v4f r4;
#pragma unroll
  for (int e = 0; e < kVec; ++e) {
    const float x  = x4[e];
    const float y  = y4[e];
    const float w  = w4[e];
    const float h  = h4[e];
    const int   pw = p4[e];

    float ox0, ox1, oy0, oy1;
    int ix0, ix1, iy0, iy1;
    axis_overlap(x, w, ox0, ox1, ix0, ix1);
    axis_overlap(y, h, oy0, oy1, iy0, iy1);

    // 4 gathers into the L2-resident map (default RT temporal policy).
    const int row0 = ix0 * kNB;
    const int row1 = ix1 * kNB;
    const float u00 = util[row0 + iy0];
    const float u01 = util[row0 + iy1];
    const float u10 = util[row1 + iy0];
    const float u11 = util[row1 + iy1];

    const float area = ox0 * (oy0 * u00 + oy1 * u01)
                     + ox1 * (oy0 * u10 + oy1 * u11);

    // area * pw / (w*h*0.1): v_rcp_f32 + mul instead of a full
    // correctly-rounded divide (~2e-7 rel. error, no index decision).
    const float denom = w * h * kUnitPinCap;
    r4[e] = (area * (float)pw) * __builtin_amdgcn_rcpf(denom);
  }

  __builtin_nontemporal_store(r4, reinterpret_cast<v4f*>(out + base));
}

extern "C" void kernel_launch(void* const* d_in, const int* in_sizes, int n_in,
                              void* d_out, int out_size, void* d_ws, size_t ws_size,
                              hipStream_t stream) {
  (void)in_sizes; (void)n_in; (void)out_size; (void)d_ws; (void)ws_size;

  const float* pos  = reinterpret_cast<const float*>(d_in[0]); // [2N]
  const float* nsx  = reinterpret_cast<const float*>(d_in[1]); // [N]
  const float* nsy  = reinterpret_cast<const float*>(d_in[2]); // [N]
  const float* util = reinterpret_cast<const float*>(d_in[3]); // [512*512]
  const int*   pw   = reinterpret_cast<const int*>(d_in[4]);   // [N]
  float*       out  = reinterpret_cast<float*>(d_out);         // [N]

  const int nvec = (kN + kVec - 1) / kVec;           // 500,000 threads
  const int grid = (nvec + kThreads - 1) / kThreads; // 1954 blocks

  hipLaunchKernelGGL(node_area_from_pinmap_kernel, dim3(grid), dim3(kThreads),
                     0, stream, pos, nsx, nsy, util, pw, out);
}